// TokenInterleaver_75436805587273
// MI455X (gfx1250) — compile-verified
//
#include <hip/hip_runtime.h>
#include <hip/hip_bf16.h>
#include <stdint.h>

// ---------------- problem constants (from reference) ----------------
#define BATCH    8
#define SEQ      4096
#define DIM      1024
#define NAUD     4
#define LAUD     512
#define NVOI     4
#define LVOI     512
#define NIMG     4
#define PIMG     256
#define LOUT     9204        // S - 12 + 4*512 + 4*512 + 4*256
#define AUDIO_ID 32001
#define VOICE_ID 32002
#define IMAGE_ID 32003

#define NT1      1024        // phase-1 block size
#define CH       (SEQ/NT1)   // 4 tokens per thread

#define WAVES_PER_BLOCK 8    // phase-2: one row per wave
#define CHUNKS    8          // 8 async b128 issues per wave = 8*512B = 4 KB

// ---------------- CDNA5 async global<->LDS helpers ----------------
// CDNA5 ISA: GLOBAL_LOAD_ASYNC_TO_LDS_B128 /
// GLOBAL_STORE_ASYNC_FROM_LDS_B128, tracked with ASYNCcnt.
// GV mode: 64-bit VGPR address + "off". LDS address is a 32-bit VGPR value.
// All data here is single-touch -> non-temporal hints (ISA §7.3) so the
// 192 MB L2 is not churned by ~600 MB of streaming traffic.
__device__ __forceinline__ void async_load16_nt(uint32_t lds_addr, const void* gaddr) {
    asm volatile("global_load_async_to_lds_b128 %0, %1, off th:TH_LOAD_NT"
                 :: "v"(lds_addr), "v"(gaddr) : "memory");
}
__device__ __forceinline__ void async_store16_nt(uint32_t lds_addr, void* gaddr) {
    asm volatile("global_store_async_from_lds_b128 %0, %1, off th:TH_STORE_NT"
                 :: "v"(gaddr), "v"(lds_addr) : "memory");
}
__device__ __forceinline__ void wait_async0() {
    asm volatile("s_wait_asynccnt 0" ::: "memory");
}

// =====================================================================
// Phase 1: build one int32 descriptor per output row.
// desc = (flatSrcRow << 3) | (mod << 1) | valid    (mod: 0=text,1=aud,2=voi,3=img)
// =====================================================================
__global__ __launch_bounds__(NT1)
void build_desc_kernel(const int* __restrict__ tid,
                       const int* __restrict__ alen,
                       const int* __restrict__ vlen,
                       int* __restrict__ desc)
{
    __shared__ int4 scanbuf[NT1];   // (cntA, cntV, cntI, -) scan
    __shared__ int  escan[NT1];     // expansion-length scan

    const int b = blockIdx.x;
    const int t = threadIdx.x;
    const int* ids = tid + b * SEQ;
    int* db = desc + b * LOUT;

    // fill all rows invalid first (tail rows beyond `total` stay 0)
    for (int o = t; o < LOUT; o += NT1) db[o] = 0;

    // pass 1: per-chunk modality tags + counts
    int tag[CH];
    int ca = 0, cv = 0, ci = 0;
#pragma unroll
    for (int k = 0; k < CH; ++k) {
        const int id = ids[t * CH + k];
        int m = 0;
        if (id == AUDIO_ID) m = 1;
        else if (id == VOICE_ID) m = 2;
        else if (id == IMAGE_ID) m = 3;
        tag[k] = m;
        ca += (m == 1); cv += (m == 2); ci += (m == 3);
    }
    scanbuf[t] = make_int4(ca, cv, ci, 0);
    __syncthreads();

    // inclusive Hillis-Steele scan of (ca,cv,ci)
    for (int off = 1; off < NT1; off <<= 1) {
        int4 add = make_int4(0, 0, 0, 0);
        if (t >= off) add = scanbuf[t - off];
        __syncthreads();
        int4 cur = scanbuf[t];
        cur.x += add.x; cur.y += add.y; cur.z += add.z;
        scanbuf[t] = cur;
        __syncthreads();
    }
    const int4 excl = (t > 0) ? scanbuf[t - 1] : make_int4(0, 0, 0, 0);
    __syncthreads();

    // pass 2: expansion length per token using global running counts
    int a_run = excl.x, v_run = excl.y, i_run = excl.z;
    int e[CH];
    int esum = 0;
#pragma unroll
    for (int k = 0; k < CH; ++k) {
        const int m = tag[k];
        int ek;
        if (m == 1)      { int idx = a_run < (NAUD - 1) ? a_run : (NAUD - 1); ek = alen[b * NAUD + idx]; ++a_run; }
        else if (m == 2) { int idx = v_run < (NVOI - 1) ? v_run : (NVOI - 1); ek = vlen[b * NVOI + idx]; ++v_run; }
        else if (m == 3) { ek = PIMG; ++i_run; }
        else             { ek = 1; }
        e[k] = ek;
        esum += ek;
    }
    escan[t] = esum;
    __syncthreads();
    for (int off = 1; off < NT1; off <<= 1) {
        int add = (t >= off) ? escan[t - off] : 0;
        __syncthreads();
        escan[t] += add;
        __syncthreads();
    }
    int o = (t > 0) ? escan[t - 1] : 0;   // exclusive start for this chunk

    // pass 3: scatter descriptors
    a_run = excl.x; v_run = excl.y; i_run = excl.z;
    for (int k = 0; k < CH; ++k) {
        const int m = tag[k];
        const int s = t * CH + k;
        if (m == 0) {
            db[o] = ((b * SEQ + s) << 3) | 1;
            o += 1;
        } else if (m == 1) {
            const int idx = a_run < (NAUD - 1) ? a_run : (NAUD - 1); ++a_run;
            const int base = (b * NAUD + idx) * LAUD;
            const int ek = e[k];
            for (int q = 0; q < ek; ++q) db[o + q] = ((base + q) << 3) | (1 << 1) | 1;
            o += ek;
        } else if (m == 2) {
            const int idx = v_run < (NVOI - 1) ? v_run : (NVOI - 1); ++v_run;
            const int base = (b * NVOI + idx) * LVOI;
            const int ek = e[k];
            for (int q = 0; q < ek; ++q) db[o + q] = ((base + q) << 3) | (2 << 1) | 1;
            o += ek;
        } else {
            const int idx = i_run < (NIMG - 1) ? i_run : (NIMG - 1); ++i_run;
            const int base = (b * NIMG + idx) * PIMG;
            for (int q = 0; q < PIMG; ++q) db[o + q] = ((base + q) << 3) | (3 << 1) | 1;
            o += PIMG;
        }
    }
}

// =====================================================================
// Phase 2: one 4 KB output row per WAVE via async global->LDS->global DMA.
// Each wave issues 8 pipelined async b128 loads (ASYNCcnt=8), one wait,
// then 8 pipelined async stores. 8 waves/block, 32 KB LDS staging.
// No cross-wave sharing -> no barriers. NT cache hints (single-touch data).
// =====================================================================
__global__ __launch_bounds__(32 * WAVES_PER_BLOCK)
void gather_rows_kernel(const float* __restrict__ th,
                        const float* __restrict__ ah,
                        const float* __restrict__ vh,
                        const float* __restrict__ ih,
                        const int* __restrict__ desc,
                        float* __restrict__ out_val,
                        float* __restrict__ out_mask,
                        float* __restrict__ out_mod,
                        int nrows)
{
    __shared__ float buf[WAVES_PER_BLOCK][DIM];     // 32 KB staging, 4 KB per wave
    const int t = threadIdx.x;
    const int w = t >> 5;                           // wave id within block
    const int l = t & 31;                           // lane id

    const int row = blockIdx.x * WAVES_PER_BLOCK + w;
    if (row >= nrows) return;

    const uint32_t lane_lds = (uint32_t)(size_t)(&buf[w][0]) + (uint32_t)(l * 16);

    const int d     = desc[row];
    const int valid = d & 1;
    const int m     = (d >> 1) & 3;
    const int srow  = d >> 3;

    float* dst = out_val + (size_t)row * DIM;

    if (valid) {
        const float* base = (m == 0) ? th : (m == 1) ? ah : (m == 2) ? vh : ih;
        const float* src  = base + (size_t)srow * DIM;
        // 8 pipelined async loads: lane l, chunk i covers bytes [i*512 + l*16, +16)
#pragma unroll
        for (int i = 0; i < CHUNKS; ++i)
            async_load16_nt(lane_lds + (uint32_t)(i * 512), src + i * 128 + l * 4);
        wait_async0();                              // all 8 loads resident in LDS
#pragma unroll
        for (int i = 0; i < CHUNKS; ++i)
            async_store16_nt(lane_lds + (uint32_t)(i * 512), dst + i * 128 + l * 4);
        wait_async0();                              // drain before wave exit
    } else {
        const float4 z = make_float4(0.f, 0.f, 0.f, 0.f);
#pragma unroll
        for (int i = 0; i < CHUNKS; ++i)
            *(float4*)(dst + i * 128 + l * 4) = z;
    }
    if (l == 0) {
        out_mask[row] = valid ? 1.0f : 0.0f;
        out_mod[row]  = valid ? (float)m : -1.0f;
    }
}

// =====================================================================
extern "C" void kernel_launch(void* const* d_in, const int* in_sizes, int n_in,
                              void* d_out, int out_size, void* d_ws, size_t ws_size,
                              hipStream_t stream)
{
    (void)in_sizes; (void)n_in; (void)out_size; (void)ws_size;

    // setup_inputs() order:
    const float* th  = (const float*)d_in[0]; // text_hidden_states  (B,S,D)
    const int*   tid = (const int*)  d_in[1]; // text_token_ids      (B,S)
    const float* ah  = (const float*)d_in[2]; // audio_hidden_states (B,NA,LA,D)
    const int*   al  = (const int*)  d_in[3]; // audio_lengths       (B,NA)
    const float* vh  = (const float*)d_in[4]; // voice_hidden_states (B,NV,LV,D)
    const int*   vl  = (const int*)  d_in[5]; // voice_lengths       (B,NV)
    const float* ih  = (const float*)d_in[6]; // image_hidden_states (B,NI,P,D)

    // outputs concatenated flat: interleaved, attention_mask, modality_map
    float* out_val  = (float*)d_out;
    float* out_mask = out_val + (size_t)BATCH * LOUT * DIM;
    float* out_mod  = out_mask + (size_t)BATCH * LOUT;

    int* desc = (int*)d_ws;                    // B*LOUT int32 descriptors

    build_desc_kernel<<<BATCH, NT1, 0, stream>>>(tid, al, vl, desc);

    const int nrows = BATCH * LOUT;            // 73632 rows = 9204 * 8
    const int nblocks = (nrows + WAVES_PER_BLOCK - 1) / WAVES_PER_BLOCK;  // 9204
    gather_rows_kernel<<<nblocks, 32 * WAVES_PER_BLOCK, 0, stream>>>(
        th, ah, vh, ih, desc, out_val, out_mask, out_mod, nrows);
}